// simple_GCN_66700842107579
// MI455X (gfx1250) — compile-verified
//
#include <hip/hip_runtime.h>

// Problem sizes (fixed by the reference)
#define N_NODES 10000
#define N_EDGES 160000
#define F_IN    128
#define G_GRAPH 64
#define NCLS    10
#define H1      1000
#define H2      700
#define H3      200
// Padded feature widths (multiples of 64 so the WMMA GEMM has no tail)
#define H1P     1024
#define H2P     704
#define H3P     256

typedef float v2f __attribute__((ext_vector_type(2)));
typedef float v8f __attribute__((ext_vector_type(8)));

// ---------------------------------------------------------------- utilities

__global__ void fill_zero(float* __restrict__ p, int n) {
  int i = blockIdx.x * blockDim.x + threadIdx.x;
  if (i < n) p[i] = 0.0f;
}

__global__ void deg_count(const int* __restrict__ dst, float* __restrict__ deg) {
  int e = blockIdx.x * blockDim.x + threadIdx.x;
  if (e < N_EDGES) atomicAdd(&deg[dst[e]], 1.0f);
}

__global__ void deg_finish(float* __restrict__ dinv) {
  int i = blockIdx.x * blockDim.x + threadIdx.x;
  if (i < N_NODES) dinv[i] = rsqrtf(dinv[i] + 1.0f);
}

// Copy W[fi x fo] into zero-padded, K-PAIR-PACKED layout:
//   Wp[(k/2) * fop * 2 + col * 2 + (k&1)] = W[k][col]   (0 outside fi x fo)
// so the WMMA B fragment {B[ka][col], B[ka+1][col]} is one aligned b64 load.
__global__ void pad_weight_pairs(const float* __restrict__ W, float* __restrict__ Wp,
                                 int fi, int fo, int fop, int total) {
  int idx = blockIdx.x * blockDim.x + threadIdx.x;
  if (idx >= total) return;
  int pairstride = fop * 2;
  int k2  = idx / pairstride;
  int rem = idx % pairstride;
  int col = rem >> 1;
  int j   = rem & 1;
  int k   = 2 * k2 + j;
  Wp[idx] = (k < fi && col < fo) ? W[k * fo + col] : 0.0f;
}

// ------------------------------------------------------- FP32 WMMA GEMM core
// C[M x Np] = A[M x Kp] @ B[Kp x Np], A row-major, B in K-pair-packed layout.
// Np % 64 == 0, Kp % 4 == 0. One wave computes a 16x64 strip (4 accumulators
// sharing the A operand). Layouts per CDNA5 ISA 7.12.2:
//   A 16x4 f32: lanes 0-15 -> M, half-wave selects K pair {0,1}/{2,3} in 2 VGPRs
//   B 4x16 f32: lanes -> N, half-wave selects K pair in 2 VGPRs
//   C/D 16x16 f32: VGPR r, lanes 0-15 -> M=r, lanes 16-31 -> M=8+r
__global__ __launch_bounds__(256) void gemm_wmma_f32(
    const float* __restrict__ A, const float* __restrict__ Bp,
    float* __restrict__ C, int Mtiles, int Kp, int Np) {
  const int tid  = threadIdx.x;
  const int wave = tid >> 5;
  const int lane = tid & 31;
  const int l    = lane & 15;
  const int half = lane >> 4;

  const int mtile = blockIdx.y * 8 + wave;
  if (mtile >= Mtiles) return;          // no barriers below: early-exit is safe
  const int m0 = mtile * 16;
  const int n0 = blockIdx.x * 64;
  const int col = n0 + l;

  // A fragment pointer: row m0+l, pair (k/2 + half); bump by 2 pairs per step
  const v2f* ap = (const v2f*)(A + (size_t)(m0 + l) * Kp) + half;
  // B fragment pointer: packed pairs, row stride Np (in v2f units)
  const v2f* bp = (const v2f*)Bp + (size_t)half * Np + col;
  const size_t bstep = (size_t)2 * Np;

  v8f acc0 = {}, acc1 = {}, acc2 = {}, acc3 = {};

  #pragma unroll 2
  for (int k = 0; k < Kp; k += 4) {
    v2f a  = *ap;      ap += 2;
    v2f b0 = bp[0];
    v2f b1 = bp[16];
    v2f b2 = bp[32];
    v2f b3 = bp[48];   bp += bstep;

    acc0 = __builtin_amdgcn_wmma_f32_16x16x4_f32(false, a, false, b0, (short)0, acc0, false, false);
    acc1 = __builtin_amdgcn_wmma_f32_16x16x4_f32(false, a, false, b1, (short)0, acc1, false, false);
    acc2 = __builtin_amdgcn_wmma_f32_16x16x4_f32(false, a, false, b2, (short)0, acc2, false, false);
    acc3 = __builtin_amdgcn_wmma_f32_16x16x4_f32(false, a, false, b3, (short)0, acc3, false, false);
  }

  // D layout: VGPR r holds row (m0 + half*8 + r), column = n0 + 16*tile + l
  float* Crow = C + (size_t)(m0 + half * 8) * Np + n0 + l;
  #pragma unroll
  for (int r = 0; r < 8; ++r) {
    Crow[(size_t)r * Np +  0] = acc0[r];
    Crow[(size_t)r * Np + 16] = acc1[r];
    Crow[(size_t)r * Np + 32] = acc2[r];
    Crow[(size_t)r * Np + 48] = acc3[r];
  }
}

// ------------------------------------------------------- GCN aggregation ops

// out[i][j] = h[i][j] * dinv[i]^2 for j<Nout, 0 in pad columns
__global__ void self_init(const float* __restrict__ h, float* __restrict__ out,
                          const float* __restrict__ dinv, int Nout, int Np, int total) {
  int idx = blockIdx.x * blockDim.x + threadIdx.x;
  if (idx >= total) return;
  int i = idx / Np, j = idx % Np;
  float v = 0.0f;
  if (j < Nout) {
    float di = dinv[i];
    v = h[idx] * di * di;
  }
  out[idx] = v;
}

// wave per edge: out[dst][j] += dinv[src]*dinv[dst]*h[src][j]
__global__ void edge_scatter(const float* __restrict__ h, float* __restrict__ out,
                             const int* __restrict__ src, const int* __restrict__ dst,
                             const float* __restrict__ dinv, int Nout, int Np) {
  int gid  = blockIdx.x * blockDim.x + threadIdx.x;
  int e    = gid >> 5;
  int lane = gid & 31;
  if (e >= N_EDGES) return;
  int s = src[e], d = dst[e];
  float c = dinv[s] * dinv[d];
  const float* hs = h + (size_t)s * Np;
  float*       od = out + (size_t)d * Np;
  for (int j = lane; j < Nout; j += 32) atomicAdd(&od[j], c * hs[j]);
}

// out[i][j] = relu(out[i][j] + b[j]) for j<Nout, keep pad columns at 0
__global__ void bias_relu(float* __restrict__ out, const float* __restrict__ b,
                          int Nout, int Np, int total) {
  int idx = blockIdx.x * blockDim.x + threadIdx.x;
  if (idx >= total) return;
  int j = idx % Np;
  out[idx] = (j < Nout) ? fmaxf(out[idx] + b[j], 0.0f) : 0.0f;
}

// wave per node: pooled[batch[i]][j] += h[i][j]; cnt[batch[i]] += 1
__global__ void pool_scatter(const float* __restrict__ h, const int* __restrict__ batch,
                             float* __restrict__ pooled, float* __restrict__ cnt) {
  int gid  = blockIdx.x * blockDim.x + threadIdx.x;
  int node = gid >> 5;
  int lane = gid & 31;
  if (node >= N_NODES) return;
  int g = batch[node];
  const float* hr = h + (size_t)node * H3P;
  float*       pr = pooled + (size_t)g * H3P;
  for (int j = lane; j < H3; j += 32) atomicAdd(&pr[j], hr[j]);
  if (lane == 0) atomicAdd(&cnt[g], 1.0f);
}

// out[g][c] = (pooled[g][:] @ Wl[:,c]) / max(cnt[g],1) + bl[c]
__global__ void final_linear(const float* __restrict__ pooled, const float* __restrict__ cnt,
                             const float* __restrict__ Wl, const float* __restrict__ bl,
                             float* __restrict__ out) {
  int t = blockIdx.x * blockDim.x + threadIdx.x;
  if (t >= G_GRAPH * NCLS) return;
  int g = t / NCLS, c = t % NCLS;
  const float* pg = pooled + (size_t)g * H3P;
  float acc = 0.0f;
  for (int j = 0; j < H3; ++j) acc += pg[j] * Wl[j * NCLS + c];
  out[t] = acc / fmaxf(cnt[g], 1.0f) + bl[c];
}

// ---------------------------------------------------------------- launcher

extern "C" void kernel_launch(void* const* d_in, const int* in_sizes, int n_in,
                              void* d_out, int out_size, void* d_ws, size_t ws_size,
                              hipStream_t stream) {
  (void)in_sizes; (void)n_in; (void)out_size; (void)ws_size;
  const float* x     = (const float*)d_in[0];
  const int*   ei    = (const int*)  d_in[1];
  const int*   batch = (const int*)  d_in[2];
  const float* W1 = (const float*)d_in[3];  const float* b1 = (const float*)d_in[4];
  const float* W2 = (const float*)d_in[5];  const float* b2 = (const float*)d_in[6];
  const float* W3 = (const float*)d_in[7];  const float* b3 = (const float*)d_in[8];
  const float* Wl = (const float*)d_in[9];  const float* bl = (const float*)d_in[10];
  float* out = (float*)d_out;
  const int* src = ei;            // edge_index[0]
  const int* dst = ei + N_EDGES;  // edge_index[1]

  char* ws = (char*)d_ws;
  size_t off = 0;
  auto take = [&](size_t bytes) -> float* {
    off = (off + 255) & ~(size_t)255;
    float* p = (float*)(ws + off);
    off += bytes;
    return p;
  };
  float* dinv   = take((size_t)N_NODES * 4);
  float* cnt    = take((size_t)G_GRAPH * 4);
  float* pooled = take((size_t)G_GRAPH * H3P * 4);
  float* W1p    = take((size_t)F_IN * H1P * 4);
  float* W2p    = take((size_t)H1P  * H2P * 4);
  float* W3p    = take((size_t)H2P  * H3P * 4);
  float* bufA   = take((size_t)N_NODES * H1P * 4);
  float* bufB   = take((size_t)N_NODES * H1P * 4);

  const int T = 256;
  const int MT = (N_NODES + 15) / 16;            // 625 M-tiles
  const int MGRID = (MT + 7) / 8;                // 8 waves (M-tiles) per block

  // degrees -> dinv = rsqrt(deg+1)
  fill_zero<<<(N_NODES + T - 1) / T, T, 0, stream>>>(dinv, N_NODES);
  deg_count<<<(N_EDGES + T - 1) / T, T, 0, stream>>>(dst, dinv);
  deg_finish<<<(N_NODES + T - 1) / T, T, 0, stream>>>(dinv);

  // zero-padded, K-pair-packed weights
  pad_weight_pairs<<<((F_IN * H1P) + T - 1) / T, T, 0, stream>>>(W1, W1p, F_IN, H1, H1P, F_IN * H1P);
  pad_weight_pairs<<<((H1P * H2P) + T - 1) / T, T, 0, stream>>>(W2, W2p, H1,  H2, H2P, H1P * H2P);
  pad_weight_pairs<<<((H2P * H3P) + T - 1) / T, T, 0, stream>>>(W3, W3p, H2,  H3, H3P, H2P * H3P);

  // ---- layer 1: h1 = x @ W1 (K=128) ; aggregate ; relu
  gemm_wmma_f32<<<dim3(H1P / 64, MGRID), T, 0, stream>>>(x, W1p, bufA, MT, F_IN, H1P);
  self_init   <<<((N_NODES * H1P) + T - 1) / T, T, 0, stream>>>(bufA, bufB, dinv, H1, H1P, N_NODES * H1P);
  edge_scatter<<<((N_EDGES * 32) + T - 1) / T, T, 0, stream>>>(bufA, bufB, src, dst, dinv, H1, H1P);
  bias_relu   <<<((N_NODES * H1P) + T - 1) / T, T, 0, stream>>>(bufB, b1, H1, H1P, N_NODES * H1P);

  // ---- layer 2: K = H1P (pad cols of bufB are zero, pad rows of W2p are zero)
  gemm_wmma_f32<<<dim3(H2P / 64, MGRID), T, 0, stream>>>(bufB, W2p, bufA, MT, H1P, H2P);
  self_init   <<<((N_NODES * H2P) + T - 1) / T, T, 0, stream>>>(bufA, bufB, dinv, H2, H2P, N_NODES * H2P);
  edge_scatter<<<((N_EDGES * 32) + T - 1) / T, T, 0, stream>>>(bufA, bufB, src, dst, dinv, H2, H2P);
  bias_relu   <<<((N_NODES * H2P) + T - 1) / T, T, 0, stream>>>(bufB, b2, H2, H2P, N_NODES * H2P);

  // ---- layer 3
  gemm_wmma_f32<<<dim3(H3P / 64, MGRID), T, 0, stream>>>(bufB, W3p, bufA, MT, H2P, H3P);
  self_init   <<<((N_NODES * H3P) + T - 1) / T, T, 0, stream>>>(bufA, bufB, dinv, H3, H3P, N_NODES * H3P);
  edge_scatter<<<((N_EDGES * 32) + T - 1) / T, T, 0, stream>>>(bufA, bufB, src, dst, dinv, H3, H3P);
  bias_relu   <<<((N_NODES * H3P) + T - 1) / T, T, 0, stream>>>(bufB, b3, H3, H3P, N_NODES * H3P);

  // ---- global mean pool + classifier head
  fill_zero<<<1, T, 0, stream>>>(cnt, G_GRAPH);
  fill_zero<<<((G_GRAPH * H3P) + T - 1) / T, T, 0, stream>>>(pooled, G_GRAPH * H3P);
  pool_scatter<<<((N_NODES * 32) + T - 1) / T, T, 0, stream>>>(bufB, batch, pooled, cnt);
  final_linear<<<((G_GRAPH * NCLS) + T - 1) / T, T, 0, stream>>>(pooled, cnt, Wl, bl, out);
}